// MaxPoolBlock_5669356831307
// MI455X (gfx1250) — compile-verified
//
#include <hip/hip_runtime.h>
#include <hip/hip_bf16.h>
#include <cfloat>

// MaxPool gather: out[p, :] = max over j of x_pad[pools[p, j], :]
// N1 = 262144 source rows, N2 = 65536 pools, MAX_NUM = 32, D = 128 (fp32).
// Gather-bound: ~1 GB of row gathers, all L2-resident (x = 128 MB < 192 MB L2).
// CDNA5 path: async global->LDS gather pipeline (ASYNCcnt), ds_load_b128 consume,
// NT stores for the streamed output.

typedef __attribute__((ext_vector_type(4))) float v4f;

#define D_DIM      128
#define MAX_NUM    32
#define N1_PAD     262144u
#define WAVES_PB   8          // 8 wave32 per block (256 threads)
#define CHUNK      8          // rows per pipeline stage
#define ROW_V4     32         // 128 floats = 32 float4 per row (one per lane)

// LDS: per wave, 2 buffers x 8 rows x 512B = 8 KB; block total = 64 KB.
#define WAVE_V4    (2 * CHUNK * ROW_V4)   // 512 float4 per wave

__global__ __launch_bounds__(256, 2) void maxpool_gather_kernel(
    const float* __restrict__ x,
    const int*   __restrict__ pools,
    float*       __restrict__ out,
    int n2)
{
    __shared__ v4f sbuf[WAVES_PB * WAVE_V4];   // 64 KB

    const int tid  = (int)threadIdx.x;
    const int lane = tid & 31;
    const int wave = tid >> 5;
    const int p    = (int)blockIdx.x * WAVES_PB + wave;
    if (p >= n2) return;   // wave-uniform exit

    // each lane holds one neighbor index for this pool
    const unsigned myidx = (unsigned)pools[(size_t)p * MAX_NUM + lane];

    // LDS byte offset of this wave's staging area:
    // low 32 bits of a flat pointer into __shared__ == AS3 offset (aperture encoding)
    v4f* bufbase = &sbuf[wave * WAVE_V4];
    const unsigned lds_base = (unsigned)(unsigned long long)(void*)bufbase;

    v4f acc;
    acc.x = -FLT_MAX; acc.y = -FLT_MAX; acc.z = -FLT_MAX; acc.w = -FLT_MAX;

    // Issue one chunk (8 rows) of async gathers into LDS buffer `buf`.
    auto issue_chunk = [&](int c, int buf) {
#pragma unroll
        for (int r = 0; r < CHUNK; ++r) {
            const int j = c * CHUNK + r;
            unsigned ij   = (unsigned)__builtin_amdgcn_readlane((int)myidx, j);
            unsigned sidx = (ij >= N1_PAD) ? 0u : ij;       // padded -> safe addr, masked later
            unsigned goff = sidx * (D_DIM * 4u) + (unsigned)lane * 16u;  // byte offset < 128 MB
            unsigned laddr = lds_base + (unsigned)((buf * CHUNK + r) * ROW_V4 + lane) * 16u;
            asm volatile("global_load_async_to_lds_b128 %0, %1, %2"
                         :: "v"(laddr), "v"(goff), "s"(x)
                         : "memory");
        }
    };

    // Consume one chunk from LDS, max-accumulating; padded rows contribute 0.0.
    auto consume_chunk = [&](int c, int buf) {
#pragma unroll
        for (int r = 0; r < CHUNK; ++r) {
            const int j = c * CHUNK + r;
            unsigned ij = (unsigned)__builtin_amdgcn_readlane((int)myidx, j);
            v4f v = sbuf[wave * WAVE_V4 + (buf * CHUNK + r) * ROW_V4 + lane];
            if (ij >= N1_PAD) { v.x = 0.f; v.y = 0.f; v.z = 0.f; v.w = 0.f; }
            acc.x = fmaxf(acc.x, v.x);
            acc.y = fmaxf(acc.y, v.y);
            acc.z = fmaxf(acc.z, v.z);
            acc.w = fmaxf(acc.w, v.w);
        }
    };

    // Software pipeline over 4 chunks with a 2-deep LDS double buffer.
    // ASYNCcnt completes in issue order, so "cnt <= 8" == oldest chunk landed.
    issue_chunk(0, 0);
    issue_chunk(1, 1);

    asm volatile("s_wait_asynccnt 8" ::: "memory");   // chunk0 in LDS
    consume_chunk(0, 0);
    asm volatile("s_wait_dscnt 0" ::: "memory");      // ds reads of buf0 drained
    issue_chunk(2, 0);

    asm volatile("s_wait_asynccnt 8" ::: "memory");   // chunk1 in LDS
    consume_chunk(1, 1);
    asm volatile("s_wait_dscnt 0" ::: "memory");      // ds reads of buf1 drained
    issue_chunk(3, 1);

    asm volatile("s_wait_asynccnt 8" ::: "memory");   // chunk2 in LDS
    consume_chunk(2, 0);

    asm volatile("s_wait_asynccnt 0" ::: "memory");   // chunk3 in LDS
    consume_chunk(3, 1);

    // Streamed output: non-temporal 128-bit store (keep L2 for x).
    v4f* orow = (v4f*)(out + (size_t)p * D_DIM);
    __builtin_nontemporal_store(acc, orow + lane);
}

extern "C" void kernel_launch(void* const* d_in, const int* in_sizes, int n_in,
                              void* d_out, int out_size, void* d_ws, size_t ws_size,
                              hipStream_t stream) {
    const float* x     = (const float*)d_in[0];
    const int*   pools = (const int*)d_in[1];   // integer input -> const int* per harness ABI
    float*       out   = (float*)d_out;

    const int n2 = out_size / D_DIM;            // 65536 pooled points
    const int blocks = (n2 + WAVES_PB - 1) / WAVES_PB;

    maxpool_gather_kernel<<<blocks, WAVES_PB * 32, 0, stream>>>(x, pools, out, n2);
}